// SLModel_42949672961096
// MI455X (gfx1250) — compile-verified
//
#include <hip/hip_runtime.h>
#include <math.h>

#define BB 64
#define SS 512
#define TT 50        // real tag dim (48 + START + END)
#define TP 64        // padded tag dim for WMMA tiles
#define LSTR 65      // LDS row stride (bank-conflict padding)
#define START_TAG 48
#define END_TAG 49

typedef __attribute__((ext_vector_type(2))) float v2f;
typedef __attribute__((ext_vector_type(8))) float v8f;

__global__ void crf_zero(float* out) { out[0] = 0.0f; }

// One workgroup per 16-batch M-tile; 4 waves, wave w owns N-columns [16w, 16w+16).
__global__ __launch_bounds__(128) void crf_forward(const float* __restrict__ feats,
                                                   const float* __restrict__ trans,
                                                   const int* __restrict__ mask,
                                                   float* __restrict__ out) {
  __shared__ float sT[TP * LSTR];   // exp(transitions), zero-padded to 64x64
  __shared__ float sP[16 * LSTR];   // partition (log space), padding = -inf
  __shared__ float sE[16 * LSTR];   // exp(partition - rowmax)
  __shared__ float sMx[8 * 16];     // partial row maxes
  __shared__ float sRm[16];         // row maxes
  __shared__ float sFin[16];        // per-batch final LSE

  const int tid  = threadIdx.x;
  const int lane = tid & 31;
  const int wv   = tid >> 5;
  const int hl   = lane >> 4;   // half-wave select
  const int ln16 = lane & 15;
  const int b0   = blockIdx.x * 16;
  const int j0   = wv * 16;

  // exp(transitions) into LDS, padded with zeros (exp(-1000) underflows to 0 anyway)
  for (int idx = tid; idx < TP * TP; idx += 128) {
    int i = idx >> 6, j = idx & 63;
    float v = 0.0f;
    if (i < TT && j < TT) v = expf(trans[i * TT + j]);
    sT[i * LSTR + j] = v;
  }
  // init partition: feats[b,0,:] + transitions[START,:]
  for (int idx = tid; idx < 16 * TP; idx += 128) {
    int b = idx >> 6, j = idx & 63;
    float v = -__builtin_inff();
    if (j < TT) v = feats[(size_t)(b0 + b) * SS * TT + j] + trans[START_TAG * TT + j];
    sP[b * LSTR + j] = v;
  }
  __syncthreads();

  // Hoist time-invariant B fragments (exp(transitions) columns j0..j0+15) to registers.
  // B layout (4x16 K-major tile): VGPR v, lanes0-15 -> K=v, lanes16-31 -> K=v+2.
  v2f bf[16];
#pragma unroll
  for (int kt = 0; kt < 16; ++kt) {
    int k = kt * 4 + 2 * hl;
    int n = j0 + ln16;
    v2f t;
    t.x = sT[k * LSTR + n];
    t.y = sT[(k + 1) * LSTR + n];
    bf[kt] = t;
  }

  for (int s = 1; s < SS; ++s) {
    __syncthreads();  // previous step's partition writes visible
    {
      // partial row max: 8 segments x 8 columns per row
      int b = tid & 15, seg = tid >> 4;
      const float* row = &sP[b * LSTR + seg * 8];
      float m = row[0];
#pragma unroll
      for (int jj = 1; jj < 8; ++jj) m = fmaxf(m, row[jj]);
      sMx[seg * 16 + b] = m;
    }
    __syncthreads();
    if (tid < 16) {
      float m = sMx[tid];
#pragma unroll
      for (int seg = 1; seg < 8; ++seg) m = fmaxf(m, sMx[seg * 16 + tid]);
      sRm[tid] = m;
    }
    __syncthreads();
    // exp(partition - rowmax); padded cols are -inf -> exp = 0
    for (int idx = tid; idx < 16 * TP; idx += 128) {
      int b = idx >> 6, j = idx & 63;
      sE[b * LSTR + j] = expf(sP[b * LSTR + j] - sRm[b]);
    }
    __syncthreads();

    // acc(16x16) = pexp(16x64) @ expT(64x16) via 16 chained f32 WMMAs
    v8f c = {};
#pragma unroll
    for (int kt = 0; kt < 16; ++kt) {
      int k = kt * 4 + 2 * hl;
      v2f a;
      a.x = sE[ln16 * LSTR + k];
      a.y = sE[ln16 * LSTR + k + 1];
      c = __builtin_amdgcn_wmma_f32_16x16x4_f32(false, a, false, bf[kt],
                                                (short)0, c, false, false);
    }

    // new_p = feats + rowmax + log(acc), masked update
    int j = j0 + ln16;
    if (j < TT) {
#pragma unroll
      for (int r = 0; r < 8; ++r) {
        int b = r + 8 * hl;  // D layout: VGPR r, lanes0-15 -> row r, lanes16-31 -> row r+8
        size_t fidx = (size_t)(b0 + b) * SS * TT + (size_t)s * TT + j;
        float f = feats[fidx];
        if (s + 1 < SS) __builtin_prefetch(&feats[fidx + TT], 0, 0);
        float np = f + sRm[b] + logf(c[r]);
        if (mask[(b0 + b) * SS + s]) sP[b * LSTR + j] = np;
      }
    }
  }
  __syncthreads();

  // terminal transition into END: LSE_i(partition[b,i] + trans[i,END])
  if (tid < 16) {
    int b = tid;
    float m = -__builtin_inff();
    for (int i = 0; i < TT; ++i) m = fmaxf(m, sP[b * LSTR + i] + trans[i * TT + END_TAG]);
    float sum = 0.0f;
    for (int i = 0; i < TT; ++i) sum += expf(sP[b * LSTR + i] + trans[i * TT + END_TAG] - m);
    sFin[b] = m + logf(sum);
  }
  __syncthreads();
  if (tid == 0) {
    float t = 0.0f;
    for (int b = 0; b < 16; ++b) t += sFin[b];
    atomicAdd(out, t * (1.0f / (float)BB));
  }
}

// Gold path score: one block per batch
__global__ __launch_bounds__(256) void crf_gold(const float* __restrict__ feats,
                                                const float* __restrict__ trans,
                                                const int* __restrict__ mask,
                                                const int* __restrict__ tags,
                                                float* __restrict__ out) {
  __shared__ float red[256];
  __shared__ int ired[256];
  int b = blockIdx.x;
  int tid = threadIdx.x;
  float acc = 0.0f;
  int msum = 0;
  for (int s = tid; s < SS; s += 256) {
    int tg = tags[b * SS + s];
    int pv = (s == 0) ? START_TAG : tags[b * SS + s - 1];
    int mk = mask[b * SS + s];
    float v = feats[(size_t)b * SS * TT + (size_t)s * TT + tg] + trans[pv * TT + tg];
    acc += mk ? v : 0.0f;
    msum += mk;
  }
  red[tid] = acc;
  ired[tid] = msum;
  __syncthreads();
  for (int off = 128; off > 0; off >>= 1) {
    if (tid < off) {
      red[tid] += red[tid + off];
      ired[tid] += ired[tid + off];
    }
    __syncthreads();
  }
  if (tid == 0) {
    int len = ired[0];
    int e = tags[b * SS + len - 1];
    float total = red[0] + trans[e * TT + END_TAG];
    atomicAdd(out, -total * (1.0f / (float)BB));
  }
}

extern "C" void kernel_launch(void* const* d_in, const int* in_sizes, int n_in,
                              void* d_out, int out_size, void* d_ws, size_t ws_size,
                              hipStream_t stream) {
  const float* feats = (const float*)d_in[0];
  const float* trans = (const float*)d_in[1];
  const int* mask = (const int*)d_in[2];
  const int* tags = (const int*)d_in[3];
  float* out = (float*)d_out;

  crf_zero<<<1, 1, 0, stream>>>(out);
  crf_forward<<<4, 128, 0, stream>>>(feats, trans, mask, out);
  crf_gold<<<BB, 256, 0, stream>>>(feats, trans, mask, tags, out);
}